// GNN_23210003267827
// MI455X (gfx1250) — compile-verified
//
#include <hip/hip_runtime.h>
#include <hip/hip_bf16.h>

#define GDIM 128  // feature dimension (fixed by reference)

typedef float v2f __attribute__((ext_vector_type(2)));
typedef float v8f __attribute__((ext_vector_type(8)));

// ---------------------------------------------------------------------------
// Degree / normalization kernels
// ---------------------------------------------------------------------------
__global__ void deg_init_kernel(float* __restrict__ deg, int n) {
  int i = blockIdx.x * blockDim.x + threadIdx.x;
  if (i < n) deg[i] = 1.0f;  // self-loop contributes 1 to every node's degree
}

__global__ void deg_edges_kernel(const int* __restrict__ dst, float* __restrict__ deg, int E) {
  int e = blockIdx.x * blockDim.x + threadIdx.x;
  if (e < E) {
    __hip_atomic_fetch_add(&deg[dst[e]], 1.0f, __ATOMIC_RELAXED, __HIP_MEMORY_SCOPE_AGENT);
  }
}

__global__ void dinv_kernel(float* __restrict__ deg, int n) {
  int i = blockIdx.x * blockDim.x + threadIdx.x;
  if (i < n) deg[i] = rsqrtf(deg[i]);  // deg >= 1 always (self loops)
}

// ---------------------------------------------------------------------------
// FP32 WMMA GEMM: C[M x 128] = A[M x 128] * B[128 x 128], row-major.
// One wave per 16x16 C tile. blockDim = 256 (8 waves); wave w owns column
// tile w (cols 16w..16w+15), blockIdx.x owns row tile (rows 16*bx..16*bx+15).
// K loop: 32 x V_WMMA_F32_16X16X4_F32.
//
// ISA VGPR layouts (CDNA5 ISA):
//   A 16x4 fp32 : lanes 0-15 (M=lane)  VGPR0=K0, VGPR1=K1
//                 lanes 16-31 (M=lane-16) VGPR0=K2, VGPR1=K3
//   B 4x16 fp32 : lanes 0-15 (N=lane)  VGPR0=K0 row, VGPR1=K1 row
//                 lanes 16-31 (N=lane-16) VGPR0=K2 row, VGPR1=K3 row
//   C 16x16 fp32: VGPR r -> row r (lanes 0-15) / row r+8 (lanes 16-31), N=lane%16
// ---------------------------------------------------------------------------
__global__ void gemm_wmma_f32_kernel(const float* __restrict__ A,
                                     const float* __restrict__ B,
                                     float* __restrict__ C, int M) {
  const int wave = threadIdx.x >> 5;   // 0..7 -> column tile
  const int lane = threadIdx.x & 31;
  const int half = lane >> 4;          // 0: K pair {k,k+1}; 1: K pair {k+2,k+3}
  const int lmod = lane & 15;
  const int row0 = blockIdx.x * 16;
  const int col0 = wave * 16;
  if (row0 >= M) return;

  const float* Arow = A + (size_t)(row0 + lmod) * GDIM;  // M index = lmod for both halves
  const float* Bcol = B + col0 + lmod;                   // N index = lmod for both halves

  v8f acc = {0.f, 0.f, 0.f, 0.f, 0.f, 0.f, 0.f, 0.f};

#pragma unroll 4
  for (int k = 0; k < GDIM; k += 4) {
    const int ka = k + half * 2;
    v2f a, b;
    a.x = Arow[ka + 0];
    a.y = Arow[ka + 1];
    b.x = Bcol[(size_t)(ka + 0) * GDIM];
    b.y = Bcol[(size_t)(ka + 1) * GDIM];
    acc = __builtin_amdgcn_wmma_f32_16x16x4_f32(
        /*neg_a=*/false, a, /*neg_b=*/false, b,
        /*c_mod=*/(short)0, acc, /*reuse_a=*/false, /*reuse_b=*/false);
  }

  // Store C tile: VGPR r -> row (row0 + half*8 + r), col (col0 + lmod)
  float* Cp = C + (size_t)(row0 + half * 8) * GDIM + col0 + lmod;
#pragma unroll
  for (int r = 0; r < 8; ++r) {
    Cp[(size_t)r * GDIM] = acc[r];
  }
}

// ---------------------------------------------------------------------------
// agg[i,:] = h[i,:] * dinv[i]^2 (self-loop message) [+ bias if provided]
// One thread per element.
// ---------------------------------------------------------------------------
__global__ void selfloop_init_kernel(const float* __restrict__ h,
                                     const float* __restrict__ dinv,
                                     const float* __restrict__ bias,
                                     float* __restrict__ agg, int n) {
  size_t idx = (size_t)blockIdx.x * blockDim.x + threadIdx.x;
  if (idx >= (size_t)n * GDIM) return;
  int node = (int)(idx >> 7);
  int d = (int)(idx & (GDIM - 1));
  float di = dinv[node];
  float v = h[idx] * di * di;
  if (bias) v += bias[d];
  agg[idx] = v;
}

// ---------------------------------------------------------------------------
// Edge scatter: agg[dst,:] += h[src,:] * dinv[src]*dinv[dst]
// One wave (32 lanes) per edge. The edge id is forced wave-uniform via
// readfirstlane so src/dst/dinv lookups become scalar (SMEM) loads instead of
// 32x-redundant vector loads; the feature gather is one GLOBAL_LOAD_B128 per
// lane and the accumulate is 4x GLOBAL_ATOMIC_ADD_F32 (L2-resident target).
// ---------------------------------------------------------------------------
__global__ void scatter_edges_kernel(const float* __restrict__ h,
                                     const int* __restrict__ src,
                                     const int* __restrict__ dst,
                                     const float* __restrict__ dinv,
                                     float* __restrict__ agg, int E) {
  const int wave = __builtin_amdgcn_readfirstlane(threadIdx.x >> 5);  // scalar
  const int e = blockIdx.x * 8 + wave;
  if (e >= E) return;
  const int lane = threadIdx.x & 31;
  const int s = src[e];                       // scalar load (uniform address)
  const int d = dst[e];                       // scalar load
  const float norm = dinv[s] * dinv[d];       // scalar loads + scalar mul
  const float4 v = ((const float4*)(h + (size_t)s * GDIM))[lane];
  float* ap = agg + (size_t)d * GDIM + lane * 4;
  __hip_atomic_fetch_add(ap + 0, v.x * norm, __ATOMIC_RELAXED, __HIP_MEMORY_SCOPE_AGENT);
  __hip_atomic_fetch_add(ap + 1, v.y * norm, __ATOMIC_RELAXED, __HIP_MEMORY_SCOPE_AGENT);
  __hip_atomic_fetch_add(ap + 2, v.z * norm, __ATOMIC_RELAXED, __HIP_MEMORY_SCOPE_AGENT);
  __hip_atomic_fetch_add(ap + 3, v.w * norm, __ATOMIC_RELAXED, __HIP_MEMORY_SCOPE_AGENT);
}

// ---------------------------------------------------------------------------
// out[i,d] = relu(agg[i,d] + bias[d])
// ---------------------------------------------------------------------------
__global__ void bias_relu_kernel(const float* __restrict__ agg,
                                 const float* __restrict__ bias,
                                 float* __restrict__ out, int n) {
  size_t idx = (size_t)blockIdx.x * blockDim.x + threadIdx.x;
  if (idx >= (size_t)n * GDIM) return;
  int d = (int)(idx & (GDIM - 1));
  out[idx] = fmaxf(agg[idx] + bias[d], 0.0f);
}

// ---------------------------------------------------------------------------
// Launcher
// Inputs (setup_inputs order): x[N*128] f32, edge_index[2*E] int,
//                              W1[128*128] f32, b1[128] f32, W2, b2.
// Output: out[N*128] f32.
// Workspace: dinv (N f32) | bufA (N*128 f32) | bufB (N*128 f32)  ~103 MB.
// ---------------------------------------------------------------------------
extern "C" void kernel_launch(void* const* d_in, const int* in_sizes, int n_in,
                              void* d_out, int out_size, void* d_ws, size_t ws_size,
                              hipStream_t stream) {
  const float* x  = (const float*)d_in[0];
  const int*   ei = (const int*)d_in[1];
  const float* W1 = (const float*)d_in[2];
  const float* b1 = (const float*)d_in[3];
  const float* W2 = (const float*)d_in[4];
  const float* b2 = (const float*)d_in[5];
  float* out = (float*)d_out;

  const int N = in_sizes[0] / GDIM;
  const int E = in_sizes[1] / 2;
  const int* src = ei;       // edge_index[0, :]
  const int* dst = ei + E;   // edge_index[1, :]

  float* dinv = (float*)d_ws;
  float* bufA = dinv + N;                    // 400000 B offset, 16B aligned
  float* bufB = bufA + (size_t)N * GDIM;

  const int T = 256;
  const int gN    = (N + T - 1) / T;
  const int gE    = (E + T - 1) / T;
  const int gElem = (int)(((size_t)N * GDIM + T - 1) / T);
  const int gEdgeWave = (E + 7) / 8;   // 8 edges (waves) per 256-thread block
  const int gGemm = N / 16;            // 100000/16 = 6250 row tiles

  // --- normalization coefficients dinv = deg^{-1/2} (deg includes self loop)
  deg_init_kernel<<<gN, T, 0, stream>>>(dinv, N);
  deg_edges_kernel<<<gE, T, 0, stream>>>(dst, dinv, E);
  dinv_kernel<<<gN, T, 0, stream>>>(dinv, N);

  // --- layer 1: t0 = x @ W1 ; agg1 = scatter(t0) ; h1 = relu(agg1 + b1)
  gemm_wmma_f32_kernel<<<gGemm, T, 0, stream>>>(x, W1, bufA, N);
  selfloop_init_kernel<<<gElem, T, 0, stream>>>(bufA, dinv, nullptr, bufB, N);
  scatter_edges_kernel<<<gEdgeWave, T, 0, stream>>>(bufA, src, dst, dinv, bufB, E);
  bias_relu_kernel<<<gElem, T, 0, stream>>>(bufB, b1, bufA, N);

  // --- layer 2: t1 = h1 @ W2 ; out = scatter(t1) + b2
  gemm_wmma_f32_kernel<<<gGemm, T, 0, stream>>>(bufA, W2, bufB, N);
  selfloop_init_kernel<<<gElem, T, 0, stream>>>(bufB, dinv, b2, out, N);
  scatter_edges_kernel<<<gEdgeWave, T, 0, stream>>>(bufB, src, dst, dinv, out, E);
}